// IDCT2_torch_2765958939275
// MI455X (gfx1250) — compile-verified
//
#include <hip/hip_runtime.h>
#include <math.h>

// CDNA5 / gfx1250: wave32, WMMA f32 16x16x4, async global->LDS staging.
typedef __attribute__((ext_vector_type(2))) float v2f;
typedef __attribute__((ext_vector_type(4))) float v4f;
typedef __attribute__((ext_vector_type(8))) float v8f;
typedef int v4i __attribute__((vector_size(16)));   // matches builtin param type

#define NDIM 128
#define NP   132   // padded LDS row stride (132 mod 64 == 4 -> conflict-free frags)

#if __has_builtin(__builtin_amdgcn_global_load_async_to_lds_b128)
#define HAVE_ASYNC_LDS 1
#endif

// ---------------------------------------------------------------------------
// Orthonormal IDCT (DCT-III) basis: M[n][k] = c_k * cos(pi*k*(2n+1)/(2N)),
// c_0 = sqrt(1/N), c_k = sqrt(2/N).  out = M @ X @ M^T == reference _idct_2d.
// ---------------------------------------------------------------------------
__global__ void build_idct_matrix(float* __restrict__ M) {
    int idx = blockIdx.x * blockDim.x + threadIdx.x;
    if (idx >= NDIM * NDIM) return;
    int n = idx >> 7;
    int k = idx & 127;
    double ck = (k == 0) ? sqrt(1.0 / (double)NDIM) : sqrt(2.0 / (double)NDIM);
    double ang = (M_PI / (2.0 * (double)NDIM)) * (double)k * (double)(2 * n + 1);
    M[idx] = (float)(ck * cos(ang));
}

// ---------------------------------------------------------------------------
// One 128x128 image per workgroup, 256 threads = 8 waves (wave32).
// Tile grid is 8x8 (16x16 tiles). Wave w owns a 2x4 block of tiles:
//   row tiles rt = 2*(w&3) .. +1, col tiles ct = 4*(w>>2) .. +3
// so each kk-step is 2 A-frag + 4 B-frag LDS loads feeding 8 WMMAs.
//
// Pass 1: T = X @ M^T, accumulated in registers, then stored TRANSPOSED into
//         the X buffer (after a barrier), so pass 2's B-fragments are
//         contiguous per lane.
// Pass 2: out = M @ T, stored non-temporally to HBM.
//
// WMMA f32 fragment layouts (ISA 7.12.2):
//   A 16x4 / B 4x16: lane L -> row/col = L%16; K pair = 2*(L/16)
//   C/D 16x16: VGPR r -> row r (lanes 0-15) / r+8 (lanes 16-31), col = L%16
// With stride NP=132, fragment bank = (4*(L%16) + kb) % 64: all 32 lanes on
// distinct bank pairs -> conflict-free.
// ---------------------------------------------------------------------------
__launch_bounds__(256)
__global__ void idct2_wmma(const float* __restrict__ x,
                           const float* __restrict__ Mg,
                           float* __restrict__ out) {
    extern __shared__ float smem[];
    float* sX = smem;              // X (padded), later reused for T^t
    float* sM = smem + NDIM * NP;  // IDCT basis (padded)

    const int tid = threadIdx.x;
    const int img = blockIdx.x;
    const float* __restrict__ src = x + (size_t)img * (NDIM * NDIM);
    float* __restrict__ dst = out + (size_t)img * (NDIM * NDIM);

    // Cooperative staging into padded LDS rows, 128-bit transfers.
    // Prefer gfx1250 async global->LDS (ASYNCcnt-tracked, no VGPR roundtrip).
    {
        const v4f* __restrict__ s4 = (const v4f*)src;
        const v4f* __restrict__ g4 = (const v4f*)Mg;
        for (int i = tid; i < NDIM * (NDIM / 4); i += 256) {
            int row = i >> 5;          // 32 v4f per row
            int seg = i & 31;
#ifdef HAVE_ASYNC_LDS
            __builtin_amdgcn_global_load_async_to_lds_b128(
                (__attribute__((address_space(1))) v4i*)(s4 + i),
                (__attribute__((address_space(3))) v4i*)&sX[row * NP + 4 * seg],
                0, 0);
            __builtin_amdgcn_global_load_async_to_lds_b128(
                (__attribute__((address_space(1))) v4i*)(g4 + i),
                (__attribute__((address_space(3))) v4i*)&sM[row * NP + 4 * seg],
                0, 0);
#else
            v4f vx = __builtin_nontemporal_load(&s4[i]);  // streamed once
            *(v4f*)&sX[row * NP + 4 * seg] = vx;
            *(v4f*)&sM[row * NP + 4 * seg] = g4[i];       // keep M cache-regular
#endif
        }
    }
#ifdef HAVE_ASYNC_LDS
#if __has_builtin(__builtin_amdgcn_s_wait_asynccnt)
    __builtin_amdgcn_s_wait_asynccnt(0);
#else
    asm volatile("s_wait_asynccnt 0x0" ::: "memory");
#endif
#endif
    __syncthreads();

    const int wave = tid >> 5;
    const int lane = tid & 31;
    const int l16  = lane & 15;
    const int half = lane >> 4;

    const int rt = (wave & 3) * 2;   // 2 row tiles
    const int ct = (wave >> 2) * 4;  // 4 col tiles

    const int ar0 = (16 * rt + l16) * NP;        // A-frag row offsets
    const int ar1 = ar0 + 16 * NP;
    int boff[4];
#pragma unroll
    for (int j = 0; j < 4; ++j) boff[j] = (16 * (ct + j) + l16) * NP;

    const v8f vzero = {};
    v8f acc[2][4];

    // ---------------- Pass 1: T = X @ M^T ----------------
#pragma unroll
    for (int i = 0; i < 2; ++i)
#pragma unroll
        for (int j = 0; j < 4; ++j) acc[i][j] = vzero;

    for (int kk = 0; kk < NDIM; kk += 4) {
        const int kb = kk + 2 * half;
        v2f a0 = *(const v2f*)&sX[ar0 + kb];
        v2f a1 = *(const v2f*)&sX[ar1 + kb];
#pragma unroll
        for (int j = 0; j < 4; ++j) {
            v2f b = *(const v2f*)&sM[boff[j] + kb];   // B[k][n] = M[n][k]
            acc[0][j] = __builtin_amdgcn_wmma_f32_16x16x4_f32(
                false, a0, false, b, (short)0, acc[0][j], false, false);
            acc[1][j] = __builtin_amdgcn_wmma_f32_16x16x4_f32(
                false, a1, false, b, (short)0, acc[1][j], false, false);
        }
    }
    __syncthreads();   // all waves finished reading X

    // Store T transposed over the X buffer: T(row,col) -> sX[col*NP + row].
#pragma unroll
    for (int i = 0; i < 2; ++i)
#pragma unroll
        for (int j = 0; j < 4; ++j)
#pragma unroll
            for (int r = 0; r < 8; ++r) {
                int row = 16 * (rt + i) + r + 8 * half;
                int col = 16 * (ct + j) + l16;
                sX[col * NP + row] = acc[i][j][r];
            }
    __syncthreads();   // T^t fully materialized

    // ---------------- Pass 2: out = M @ T ----------------
#pragma unroll
    for (int i = 0; i < 2; ++i)
#pragma unroll
        for (int j = 0; j < 4; ++j) acc[i][j] = vzero;

    for (int kk = 0; kk < NDIM; kk += 4) {
        const int kb = kk + 2 * half;
        v2f a0 = *(const v2f*)&sM[ar0 + kb];
        v2f a1 = *(const v2f*)&sM[ar1 + kb];
#pragma unroll
        for (int j = 0; j < 4; ++j) {
            v2f b = *(const v2f*)&sX[boff[j] + kb];   // B[k][n] = T^t[n][k]
            acc[0][j] = __builtin_amdgcn_wmma_f32_16x16x4_f32(
                false, a0, false, b, (short)0, acc[0][j], false, false);
            acc[1][j] = __builtin_amdgcn_wmma_f32_16x16x4_f32(
                false, a1, false, b, (short)0, acc[1][j], false, false);
        }
    }

    // Stream result tiles to HBM (never re-read -> non-temporal).
#pragma unroll
    for (int i = 0; i < 2; ++i)
#pragma unroll
        for (int j = 0; j < 4; ++j)
#pragma unroll
            for (int r = 0; r < 8; ++r) {
                int row = 16 * (rt + i) + r + 8 * half;
                int col = 16 * (ct + j) + l16;
                __builtin_nontemporal_store(acc[i][j][r], &dst[row * NDIM + col]);
            }
}

extern "C" void kernel_launch(void* const* d_in, const int* in_sizes, int n_in,
                              void* d_out, int out_size, void* d_ws, size_t ws_size,
                              hipStream_t stream) {
    const float* x = (const float*)d_in[0];
    float* outp = (float*)d_out;
    float* M = (float*)d_ws;   // 64 KB basis matrix in workspace

    hipLaunchKernelGGL(build_idct_matrix, dim3((NDIM * NDIM) / 256), dim3(256),
                       0, stream, M);

    const int nimg = in_sizes[0] / (NDIM * NDIM);            // 3072
    const size_t lds_bytes = 2u * NDIM * NP * sizeof(float); // 132 KB
    hipLaunchKernelGGL(idct2_wmma, dim3(nimg), dim3(256), lds_bytes, stream,
                       x, M, outp);
}